// GATForMultipleChoice_18073222381706
// MI455X (gfx1250) — compile-verified
//
#include <hip/hip_runtime.h>
#include <math.h>

#define N_NODES 10000
#define N_EDGES 128000
#define N_ANS   5
#define DIM     300
#define H1      8
#define C1      64
#define HID     512
#define NEG_SLOPE 0.2f
#define EPSF 1e-16f

typedef __attribute__((ext_vector_type(2))) float v2f;
typedef __attribute__((ext_vector_type(8))) float v8f;

__device__ __forceinline__ float lrelu(float x) { return x > 0.f ? x : NEG_SLOPE * x; }

// Correct float atomic max via sign-split int punning (init value must be -inf).
__device__ __forceinline__ void atomicMaxF(float* addr, float val) {
    if (val >= 0.f)
        atomicMax((int*)addr, __float_as_int(val));
    else
        atomicMin((unsigned int*)addr, __float_as_uint(val));
}

__global__ void fill_kernel(float* __restrict__ p, float v, int n) {
    int i = blockIdx.x * blockDim.x + threadIdx.x;
    if (i < n) p[i] = v;
}

// h = x(N,300) @ W(300,512) via V_WMMA_F32_16X16X4_F32.
// One wave -> 16x64 output strip (4 accumulators, A reused).
__global__ void gemm1_wmma(const float* __restrict__ x, const float* __restrict__ W,
                           float* __restrict__ h) {
    int wave  = (int)((blockIdx.x * blockDim.x + threadIdx.x) >> 5);
    int lane  = threadIdx.x & 31;
    int tileM = wave >> 3;          // 0..624
    int cg    = wave & 7;           // 64-col group
    if (tileM >= N_NODES / 16) return;
    int half = lane >> 4;           // 0 or 1
    int l15  = lane & 15;

    v8f acc0 = {}, acc1 = {}, acc2 = {}, acc3 = {};
    const float* xrow = x + (size_t)(tileM * 16 + l15) * DIM;
    int nbase = cg * 64 + l15;

    for (int k = 0; k < DIM; k += 4) {
        int ka = k + half * 2;
        v2f a;
        a.x = xrow[ka];
        a.y = xrow[ka + 1];
        const float* w0 = W + (size_t)ka * HID + nbase;
        const float* w1 = W + (size_t)(ka + 1) * HID + nbase;
        v2f b;
        b.x = w0[0];  b.y = w1[0];
        acc0 = __builtin_amdgcn_wmma_f32_16x16x4_f32(false, a, false, b, (short)0, acc0, false, false);
        b.x = w0[16]; b.y = w1[16];
        acc1 = __builtin_amdgcn_wmma_f32_16x16x4_f32(false, a, false, b, (short)0, acc1, false, false);
        b.x = w0[32]; b.y = w1[32];
        acc2 = __builtin_amdgcn_wmma_f32_16x16x4_f32(false, a, false, b, (short)0, acc2, false, false);
        b.x = w0[48]; b.y = w1[48];
        acc3 = __builtin_amdgcn_wmma_f32_16x16x4_f32(false, a, false, b, (short)0, acc3, false, false);
    }

    int rowbase = tileM * 16 + half * 8;
    int colbase = cg * 64 + l15;
#pragma unroll
    for (int r = 0; r < 8; r++) {
        float* hp = h + (size_t)(rowbase + r) * HID + colbase;
        hp[0]  = acc0[r];
        hp[16] = acc1[r];
        hp[32] = acc2[r];
        hp[48] = acc3[r];
    }
}

// Per (node, head): alpha_src / alpha_dst dot products over 64 channels.
__global__ void node1_alpha(const float* __restrict__ h, const float* __restrict__ ws_,
                            const float* __restrict__ wd_, float* __restrict__ asrc,
                            float* __restrict__ adst) {
    int idx = blockIdx.x * blockDim.x + threadIdx.x;
    if (idx >= N_NODES * H1) return;
    int hd = idx & 7, n = idx >> 3;
    const float* hp = h + (size_t)n * HID + hd * C1;
    const float* ws = ws_ + hd * C1;
    const float* wd = wd_ + hd * C1;
    float s = 0.f, d = 0.f;
#pragma unroll 4
    for (int c = 0; c < C1; c++) {
        float v = hp[c];
        s += v * ws[c];
        d += v * wd[c];
    }
    asrc[idx] = s;
    adst[idx] = d;
}

__global__ void edge1_max(const int* __restrict__ src, const int* __restrict__ dst,
                          const float* __restrict__ asrc, const float* __restrict__ adst,
                          float* __restrict__ ebuf, float* __restrict__ m) {
    int idx = blockIdx.x * blockDim.x + threadIdx.x;
    if (idx >= N_EDGES * H1) return;
    int hd = idx & 7, e = idx >> 3;
    int s = src[e], d = dst[e];
    float v = lrelu(asrc[s * H1 + hd] + adst[d * H1 + hd]);
    ebuf[idx] = v;
    atomicMaxF(&m[d * H1 + hd], v);
}

__global__ void edge1_exp(const int* __restrict__ dst, float* __restrict__ ebuf,
                          const float* __restrict__ m, float* __restrict__ den) {
    int idx = blockIdx.x * blockDim.x + threadIdx.x;
    if (idx >= N_EDGES * H1) return;
    int hd = idx & 7, e = idx >> 3;
    int d = dst[e];
    float ex = expf(ebuf[idx] - m[d * H1 + hd]);
    ebuf[idx] = ex;
    atomicAdd(&den[d * H1 + hd], ex);
}

// Scatter messages: idx over E*512, channel fastest for coalescing.
__global__ void edge1_agg(const int* __restrict__ src, const int* __restrict__ dst,
                          const float* __restrict__ h, const float* __restrict__ ebuf,
                          const float* __restrict__ den, float* __restrict__ out) {
    int idx = blockIdx.x * blockDim.x + threadIdx.x;
    if (idx >= N_EDGES * HID) return;
    int e = idx >> 9;
    int r = idx & 511;
    int hd = r >> 6;
    int s = src[e], d = dst[e];
    float alpha = ebuf[e * H1 + hd] / (den[d * H1 + hd] + EPSF);
    atomicAdd(&out[(size_t)d * HID + r], alpha * h[(size_t)s * HID + r]);
}

__global__ void relu_bias(float* __restrict__ out, const float* __restrict__ bias) {
    int idx = blockIdx.x * blockDim.x + threadIdx.x;
    if (idx >= N_NODES * HID) return;
    out[idx] = fmaxf(out[idx] + bias[idx & 511], 0.f);
}

// Wave-per-node GEMV: t = out1[n,:] . W2 ; then scalar attn logits.
__global__ void node2_proj(const float* __restrict__ out1, const float* __restrict__ W2,
                           const float* __restrict__ as2w, const float* __restrict__ ad2w,
                           float* __restrict__ t2, float* __restrict__ as2,
                           float* __restrict__ ad2) {
    int wave = (int)((blockIdx.x * blockDim.x + threadIdx.x) >> 5);
    int lane = threadIdx.x & 31;
    if (wave >= N_NODES) return;
    const float* row = out1 + (size_t)wave * HID;
    float s = 0.f;
    for (int i = lane; i < HID; i += 32) s += row[i] * W2[i];
#pragma unroll
    for (int off = 16; off; off >>= 1) s += __shfl_xor(s, off, 32);
    if (lane == 0) {
        t2[wave]  = s;
        as2[wave] = s * as2w[0];
        ad2[wave] = s * ad2w[0];
    }
}

__global__ void edge2_max(const int* __restrict__ src, const int* __restrict__ dst,
                          const float* __restrict__ as2, const float* __restrict__ ad2,
                          float* __restrict__ ebuf, float* __restrict__ m) {
    int e = blockIdx.x * blockDim.x + threadIdx.x;
    if (e >= N_EDGES) return;
    float v = lrelu(as2[src[e]] + ad2[dst[e]]);
    ebuf[e] = v;
    atomicMaxF(&m[dst[e]], v);
}

__global__ void edge2_exp(const int* __restrict__ dst, float* __restrict__ ebuf,
                          const float* __restrict__ m, float* __restrict__ den) {
    int e = blockIdx.x * blockDim.x + threadIdx.x;
    if (e >= N_EDGES) return;
    int d = dst[e];
    float ex = expf(ebuf[e] - m[d]);
    ebuf[e] = ex;
    atomicAdd(&den[d], ex);
}

__global__ void edge2_agg(const int* __restrict__ src, const int* __restrict__ dst,
                          const float* __restrict__ ebuf, const float* __restrict__ den,
                          const float* __restrict__ t2, float* __restrict__ out2,
                          float* __restrict__ att_row) {
    int e = blockIdx.x * blockDim.x + threadIdx.x;
    if (e >= N_EDGES) return;
    int d = dst[e];
    float alpha = ebuf[e] / (den[d] + EPSF);
    atomicAdd(&out2[d], alpha * t2[src[e]]);
    atomicAdd(&att_row[d], alpha);
}

__global__ void proba_reduce(const float* __restrict__ out2, float* __restrict__ acc) {
    int i = blockIdx.x * blockDim.x + threadIdx.x;
    if (i < N_NODES) atomicAdd(acc, out2[i]);
}

__global__ void finalize(const float* __restrict__ acc, const float* __restrict__ bias2,
                         const float* __restrict__ att, float* __restrict__ out) {
    int i = blockIdx.x * blockDim.x + threadIdx.x;
    float p[N_ANS];
#pragma unroll
    for (int a = 0; a < N_ANS; a++) p[a] = acc[a] / (float)N_NODES + bias2[0];
    if (i < N_ANS) out[i] = p[i];
    int best = 0;
#pragma unroll
    for (int a = 1; a < N_ANS; a++)
        if (p[a] > p[best]) best = a;   // first-max tie-break like jnp.argmax
    if (i < N_NODES) out[N_ANS + i] = att[best * N_NODES + i];
}

extern "C" void kernel_launch(void* const* d_in, const int* in_sizes, int n_in,
                              void* d_out, int out_size, void* d_ws, size_t ws_size,
                              hipStream_t stream) {
    const float* x     = (const float*)d_in[0];  // [A,N,D]
    const int*   ei    = (const int*)d_in[1];    // [A,2,E]
    const float* W1    = (const float*)d_in[2];  // [D, H1*C1]
    const float* aS1   = (const float*)d_in[3];  // [H1,C1]
    const float* aD1   = (const float*)d_in[4];  // [H1,C1]
    const float* b1    = (const float*)d_in[5];  // [512]
    const float* W2    = (const float*)d_in[6];  // [512,1]
    const float* aS2   = (const float*)d_in[7];  // [1,1]
    const float* aD2   = (const float*)d_in[8];  // [1,1]
    const float* b2    = (const float*)d_in[9];  // [1]
    float* out = (float*)d_out;

    float* w = (float*)d_ws;
    float* h1    = w; w += (size_t)N_NODES * HID;
    float* out1  = w; w += (size_t)N_NODES * HID;
    float* asrc  = w; w += N_NODES * H1;
    float* adst  = w; w += N_NODES * H1;
    float* m1    = w; w += N_NODES * H1;
    float* den1  = w; w += N_NODES * H1;
    float* ebuf1 = w; w += (size_t)N_EDGES * H1;
    float* t2    = w; w += N_NODES;
    float* as2   = w; w += N_NODES;
    float* ad2   = w; w += N_NODES;
    float* m2    = w; w += N_NODES;
    float* den2  = w; w += N_NODES;
    float* ebuf2 = w; w += N_EDGES;
    float* out2  = w; w += N_NODES;
    float* att   = w; w += N_ANS * N_NODES;
    float* pacc  = w; w += 8;

    const int B = 256;
    const float NINF = -__builtin_inff();
    auto blocks = [](int n, int b) { return (n + b - 1) / b; };

    fill_kernel<<<blocks(N_ANS * N_NODES, B), B, 0, stream>>>(att, 0.f, N_ANS * N_NODES);
    fill_kernel<<<1, B, 0, stream>>>(pacc, 0.f, 8);

    for (int a = 0; a < N_ANS; a++) {
        const float* xa  = x + (size_t)a * N_NODES * DIM;
        const int*   sa  = ei + (size_t)a * 2 * N_EDGES;
        const int*   da  = sa + N_EDGES;
        float*       atr = att + (size_t)a * N_NODES;

        // init per-answer accumulators
        fill_kernel<<<blocks(N_NODES * HID, B), B, 0, stream>>>(out1, 0.f, N_NODES * HID);
        fill_kernel<<<blocks(N_NODES * H1, B), B, 0, stream>>>(m1, NINF, N_NODES * H1);
        fill_kernel<<<blocks(N_NODES * H1, B), B, 0, stream>>>(den1, 0.f, N_NODES * H1);
        fill_kernel<<<blocks(N_NODES, B), B, 0, stream>>>(m2, NINF, N_NODES);
        fill_kernel<<<blocks(N_NODES, B), B, 0, stream>>>(den2, 0.f, N_NODES);
        fill_kernel<<<blocks(N_NODES, B), B, 0, stream>>>(out2, 0.f, N_NODES);

        // Layer 1
        gemm1_wmma<<<625, 256, 0, stream>>>(xa, W1, h1);  // 5000 waves, 16x64 strips
        node1_alpha<<<blocks(N_NODES * H1, B), B, 0, stream>>>(h1, aS1, aD1, asrc, adst);
        edge1_max<<<blocks(N_EDGES * H1, B), B, 0, stream>>>(sa, da, asrc, adst, ebuf1, m1);
        edge1_exp<<<blocks(N_EDGES * H1, B), B, 0, stream>>>(da, ebuf1, m1, den1);
        edge1_agg<<<blocks(N_EDGES * HID, B), B, 0, stream>>>(sa, da, h1, ebuf1, den1, out1);
        relu_bias<<<blocks(N_NODES * HID, B), B, 0, stream>>>(out1, b1);

        // Layer 2
        node2_proj<<<blocks(N_NODES * 32, B), B, 0, stream>>>(out1, W2, aS2, aD2, t2, as2, ad2);
        edge2_max<<<blocks(N_EDGES, B), B, 0, stream>>>(sa, da, as2, ad2, ebuf2, m2);
        edge2_exp<<<blocks(N_EDGES, B), B, 0, stream>>>(da, ebuf2, m2, den2);
        edge2_agg<<<blocks(N_EDGES, B), B, 0, stream>>>(sa, da, ebuf2, den2, t2, out2, atr);
        proba_reduce<<<blocks(N_NODES, B), B, 0, stream>>>(out2, pacc + a);
    }

    finalize<<<blocks(N_NODES, B), B, 0, stream>>>(pacc, b2, att, out);
    (void)in_sizes; (void)n_in; (void)out_size; (void)ws_size;
}